// LengthRegulator_6244882448636
// MI455X (gfx1250) — compile-verified
//
#include <hip/hip_runtime.h>
#include <stdint.h>

#define B_  32
#define C_  384
#define T_  512
#define MF_ 6656          // MAX_FRAMES = T*13
#define CG  16            // channels per expand block (32KB LDS slab)

// ---- CDNA5 async global->LDS copy (ASYNCcnt path), GVS addressing ----
__device__ __forceinline__ void async_load_b128(uint32_t lds_addr, uint32_t goff,
                                                const void* sbase) {
  // vdst = VGPR holding LDS byte address, vaddr = 32-bit byte offset, saddr = SGPR64 base
  asm volatile("global_load_async_to_lds_b128 %0, %1, %2"
               :: "v"(lds_addr), "v"(goff), "s"(sbase)
               : "memory");
}
__device__ __forceinline__ void wait_asynccnt0() {
  asm volatile("s_wait_asynccnt 0" ::: "memory");
}

// ---------- Kernel A: repeat counts + inclusive scan (one block per batch) ----------
__global__ __launch_bounds__(T_) void lr_scan(const float* __restrict__ dur,
                                              int* __restrict__ cum,
                                              float* __restrict__ flen_out) {
  const int b = blockIdx.x, t = threadIdx.x;
  int r;
  {
    // Must match XLA op-by-op rounding: no mul+sub FMA contraction.
    #pragma clang fp contract(off)
    float d   = dur[b * T_ + t];
    float fr  = 44100.0f * d;            // rounded to f32 first
    float frm = fr - 1024.0f;            // then subtracted (separate rounding)
    float rf;
    if (frm > 0.0f) rf = fmaxf(frm * (1.0f / 256.0f), 1.0f);   // /256 exact pow2
    else            rf = (d == 0.0f) ? 0.0f : 1.0f;
    r = (int)rf;                         // astype(int32): truncate toward zero
  }

  __shared__ int s[T_];
  s[t] = r;
  __syncthreads();
  #pragma unroll
  for (int off = 1; off < T_; off <<= 1) {
    int v = (t >= off) ? s[t - off] : 0;
    __syncthreads();
    s[t] += v;
    __syncthreads();
  }
  cum[b * T_ + t] = s[t];
  if (t == T_ - 1) flen_out[b] = (float)s[t];
}

// ---------- Kernel B: searchsorted(cum, pos, 'right') + pitch ----------
__global__ __launch_bounds__(256) void lr_tok(const int*   __restrict__ cum,
                                              const float* __restrict__ notepitch,
                                              int*   __restrict__ tok,
                                              float* __restrict__ pitch) {
  const int b   = blockIdx.y;
  const int pos = blockIdx.x * 256 + threadIdx.x;
  __shared__ int cs[T_];
  cs[threadIdx.x]       = cum[b * T_ + threadIdx.x];
  cs[threadIdx.x + 256] = cum[b * T_ + threadIdx.x + 256];
  __syncthreads();

  int lo = 0, hi = T_;                   // first index with cs[idx] > pos
  while (lo < hi) { int mid = (lo + hi) >> 1; if (cs[mid] <= pos) lo = mid + 1; else hi = mid; }
  const int  flen = cs[T_ - 1];
  const int  tc   = min(lo, T_ - 1);
  const bool m    = pos < flen;

  pitch[(size_t)b * MF_ + pos] = m ? (float)(int)notepitch[b * T_ + tc] : 0.0f;
  tok  [(size_t)b * MF_ + pos] = m ? tc : -1;
}

// ---------- Kernel C: expansion; async-stage x slab in LDS, NT streaming stores ----------
__global__ __launch_bounds__(256) void lr_expand(const float* __restrict__ x,
                                                 const int*   __restrict__ tok,
                                                 float*       __restrict__ out) {
  __shared__ float xs[CG * T_];          // 32 KB contiguous slab: channels cg*16..+15
  const int b   = blockIdx.y;
  const int cg  = blockIdx.x;            // 0..23
  const int tid = threadIdx.x;

  const float* gsrc = x + ((size_t)b * C_ + (size_t)cg * CG) * T_;   // contiguous 32 KB
  const uint32_t lbase = (uint32_t)(uintptr_t)(&xs[0]);              // flat LDS -> low 32b offset
  #pragma unroll
  for (int k = 0; k < (CG * T_ * 4) / (256 * 16); ++k) {             // 8 × b128 per thread
    const uint32_t off = (uint32_t)(tid + k * 256) * 16u;
    async_load_b128(lbase + off, off, (const void*)gsrc);
  }
  wait_asynccnt0();
  __syncthreads();

  const int*  tokb = tok + (size_t)b * MF_;
  float*      outb = out + ((size_t)b * C_ + (size_t)cg * CG) * MF_;
  for (int fc = tid; fc < MF_; fc += 256) {
    const int  t  = tokb[fc];
    const bool m  = t >= 0;
    const int  ti = m ? t : 0;
    #pragma unroll
    for (int c = 0; c < CG; ++c) {
      const float v = m ? xs[c * T_ + ti] : 0.0f;
      __builtin_nontemporal_store(v, &outb[(size_t)c * MF_ + fc]);   // 327MB stream: keep L2 for x
    }
  }
}

extern "C" void kernel_launch(void* const* d_in, const int* in_sizes, int n_in,
                              void* d_out, int out_size, void* d_ws, size_t ws_size,
                              hipStream_t stream) {
  const float* x         = (const float*)d_in[0];
  const float* notepitch = (const float*)d_in[1];
  const float* duration  = (const float*)d_in[2];
  // d_in[3] (x_lengths) is unused by the reference.

  float* out   = (float*)d_out;                       // (B, C, MF)
  float* pitch = out   + (size_t)B_ * C_ * MF_;       // (B, MF)
  float* flen  = pitch + (size_t)B_ * MF_;            // (B,) as float

  int* cum = (int*)d_ws;                              // B*T ints
  int* tok = cum + B_ * T_;                           // B*MF ints  (total < 1 MB)

  lr_scan  <<<B_, T_, 0, stream>>>(duration, cum, flen);
  lr_tok   <<<dim3(MF_ / 256, B_), 256, 0, stream>>>(cum, notepitch, tok, pitch);
  lr_expand<<<dim3(C_ / CG, B_), 256, 0, stream>>>(x, tok, out);
}